// Dist2CycleLayer_4191888081073
// MI455X (gfx1250) — compile-verified
//
#include <hip/hip_runtime.h>

typedef __attribute__((ext_vector_type(4))) float v4f;
typedef __attribute__((ext_vector_type(2))) float v2f;
typedef __attribute__((ext_vector_type(8))) float v8f;

#define TN      8192      // N
#define TILE_K  256       // K columns staged per iteration
#define LSTRIDE 260       // LDS row stride (floats): 16B-aligned rows, conflict-free b64 frag loads
#define ROWS    16        // M-tile rows per block
#define NWAVES  8         // 256 threads / wave32

#if defined(__gfx1250__) && __has_builtin(__builtin_amdgcn_wmma_f32_16x16x4_f32)
#define HAVE_WMMA_F32X4 1
#else
#define HAVE_WMMA_F32X4 0
#endif

__global__ __launch_bounds__(256) void d2c_rowdot_kernel(
    const float* __restrict__ Linv,
    const float* __restrict__ adj,
    const float* __restrict__ W,
    const float* __restrict__ bias,
    float* __restrict__ out)
{
    __shared__ __align__(16) float tile[2][ROWS * LSTRIDE];
    __shared__ float partial[NWAVES * ROWS];

    const int tid      = threadIdx.x;
    const int lane     = tid & 31;          // wave32 on gfx1250
    const int wv       = tid >> 5;
    const int blockRow = blockIdx.x * ROWS;

    // A-fragment addressing for V_WMMA_F32_16X16X4_F32 (documented layout):
    // lanes 0-15 hold (K=0,K=1) of row=lane; lanes 16-31 hold (K=2,K=3) of row=lane-16.
    const int frow = lane & 15;
    const int koff = (lane < 16) ? 0 : 2;

#if HAVE_WMMA_F32X4
    v8f acc = {};                            // C/D accumulator, carries row sums
    const v2f ones = {1.0f, 1.0f};           // all-ones B => every D column == row sum
#else
    float facc = 0.0f;                       // lane-local fallback accumulator
#endif

    const int nTiles = TN / TILE_K;          // 32
    for (int t = 0; t < nTiles; ++t) {
        const int buf  = t & 1;
        const int col0 = t * TILE_K;

        // ---- Stage: s = relu(adj * Linv) * W into LDS (fused, stored once) ----
#pragma unroll
        for (int i = 0; i < 4; ++i) {
            const int f   = tid + 256 * i;            // float4 slot in 16x256 tile
            const int row = f >> 6;                   // 64 float4 per row
            const int c4  = (f & 63) << 2;            // float column within tile
            const size_t g = (size_t)(blockRow + row) * (size_t)TN + (size_t)(col0 + c4);

            v4f a = __builtin_nontemporal_load((const v4f*)(adj  + g)); // stream-once
            v4f l = __builtin_nontemporal_load((const v4f*)(Linv + g)); // stream-once
            v4f w = *(const v4f*)(W + col0 + c4);                       // L2-hot, reused
            v4f p = a * l;
            p = __builtin_elementwise_max(p, (v4f)0.0f);                // relu
            p = p * w;                                                  // pre-scale by W
            *(v4f*)(&tile[buf][row * LSTRIDE + c4]) = p;

            if (i == 0 && t + 1 < nTiles) {           // hint next tile -> global_prefetch_b8
                __builtin_prefetch(adj  + g + TILE_K, 0, 0);
                __builtin_prefetch(Linv + g + TILE_K, 0, 0);
            }
        }
        __syncthreads();

        // ---- Consume: this wave's K-share = 32 cols -> 8 WMMAs (K=4 each) ----
        const float* tb = &tile[buf][frow * LSTRIDE + wv * 32 + koff];
#pragma unroll
        for (int s = 0; s < 8; ++s) {
            v2f afrag = *(const v2f*)(tb + s * 4);    // ds_load_b64, conflict-free
#if HAVE_WMMA_F32X4
            // D = A x ones + C : accumulates per-row partial sums (replicated over N)
            acc = __builtin_amdgcn_wmma_f32_16x16x4_f32(
                false, afrag, false, ones, (short)0, acc, false, false);
#else
            facc += afrag.x + afrag.y;
#endif
        }
    }

    // ---- Per-wave row partials -> LDS (fixed order => deterministic) ----
#if HAVE_WMMA_F32X4
    // C/D layout: VGPR r, lanes 0-15 -> row r; lanes 16-31 -> row r+8 (all N equal).
    if (lane == 0) {
#pragma unroll
        for (int r = 0; r < 8; ++r) partial[wv * ROWS + r] = acc[r];
    }
    if (lane == 16) {
#pragma unroll
        for (int r = 0; r < 8; ++r) partial[wv * ROWS + 8 + r] = acc[r];
    }
#else
    {   // row m is served by lanes m and m+16
        float v = facc + __shfl_down(facc, 16, 32);
        if (lane < 16) partial[wv * ROWS + lane] = v;
    }
#endif
    __syncthreads();

    if (tid < ROWS) {
        float s = 0.0f;
#pragma unroll
        for (int w = 0; w < NWAVES; ++w) s += partial[w * ROWS + tid];
        out[blockRow + tid] = s + bias[0];
    }
}

extern "C" void kernel_launch(void* const* d_in, const int* in_sizes, int n_in,
                              void* d_out, int out_size, void* d_ws, size_t ws_size,
                              hipStream_t stream) {
    // setup_inputs order: x_e (unused), Linv, adjacency, W, b
    const float* Linv = (const float*)d_in[1];
    const float* adj  = (const float*)d_in[2];
    const float* W    = (const float*)d_in[3];
    const float* b    = (const float*)d_in[4];
    float* out = (float*)d_out;

    d2c_rowdot_kernel<<<TN / ROWS, 256, 0, stream>>>(Linv, adj, W, b, out);
}